// GCN_20882130993418
// MI455X (gfx1250) — compile-verified
//
#include <hip/hip_runtime.h>
#include <hip/hip_bf16.h>

typedef __attribute__((ext_vector_type(2))) float v2f;
typedef __attribute__((ext_vector_type(8))) float v8f;

// ---------------------------------------------------------------------------
// Utility: fill a float buffer with a constant.
// ---------------------------------------------------------------------------
__global__ void gcn_fill_kernel(float* __restrict__ p, float v, int n) {
    int i = blockIdx.x * blockDim.x + threadIdx.x;
    int stride = gridDim.x * blockDim.x;
    for (; i < n; i += stride) p[i] = v;
}

// ---------------------------------------------------------------------------
// Degree count: deg[dst[e]] += 1 (deg pre-initialized to 1.0 for self loop).
// ---------------------------------------------------------------------------
__global__ void gcn_degree_kernel(const int* __restrict__ dst,
                                  float* __restrict__ deg, int nE) {
    int e = blockIdx.x * blockDim.x + threadIdx.x;
    if (e < nE) atomicAdd(&deg[dst[e]], 1.0f);
}

// dinv[i] = rsqrt(deg[i]), in place.
__global__ void gcn_rsqrt_kernel(float* __restrict__ d, int n) {
    int i = blockIdx.x * blockDim.x + threadIdx.x;
    if (i < n) d[i] = rsqrtf(d[i]);
}

// ---------------------------------------------------------------------------
// Dense GEMM C[M,N] = A[M,K] * B[K,N] via V_WMMA_F32_16X16X4_F32.
// Compile-time K/N/NTILES -> fully unrolled, predication-free main loop.
//
//  * B is staged once per block into LDS, TRANSPOSED and ZERO-PADDED to
//    [NP x KS] (NP = NTILES*16, KS = KP+2 for bank-conflict-free b64 reads).
//    Padded columns/rows read 0.0f, so the hot loop has no N/K guards.
//  * A loads are float2 (global_load_b64): (row*K + ka)*4 is 8B-aligned for
//    K = 50 and K = 64 since ka is always even.
//  * One wave32 per 16x16 output tile. M must be a multiple of 16
//    (50000 = 3125*16). Whole-wave early exit keeps EXEC all-1s for WMMA.
//
// Lane layouts per CDNA5 ISA 7.12.2 (16x16x4 f32):
//   A: lane L -> row M = L%16 ; VGPR v holds K = 2*(L/16) + v
//   B: lane L -> col N = L%16 ; VGPR v holds K = 2*(L/16) + v
//   C/D: VGPR r -> M = r + 8*(L/16), N = L%16
// ---------------------------------------------------------------------------
template <int K, int N, int NTILES>
__global__ __launch_bounds__(256) void gcn_gemm_wmma_kernel(
    const float* __restrict__ A, const float* __restrict__ B,
    float* __restrict__ C, int M) {
    constexpr int KP = (K + 3) & ~3;   // K padded to multiple of 4
    constexpr int KS = KP + 2;         // LDS row stride (even, conflict-free)
    constexpr int NP = NTILES * 16;    // N padded to multiple of 16
    __shared__ float BT[NP * KS];

    // Cooperative staging of B^T into LDS with zero padding.
    for (int idx = threadIdx.x; idx < NP * KS; idx += blockDim.x) {
        const int n = idx / KS;
        const int k = idx - n * KS;
        BT[idx] = (n < N && k < K) ? B[(size_t)k * N + n] : 0.0f;
    }
    __syncthreads();

    const int wavesPerBlock = blockDim.x >> 5;
    const int tile = blockIdx.x * wavesPerBlock + (threadIdx.x >> 5);
    const int totalTiles = (M >> 4) * NTILES;
    if (tile >= totalTiles) return;   // whole-wave exit

    const int lane  = threadIdx.x & 31;
    const int mt    = tile / NTILES;
    const int nt    = tile - mt * NTILES;
    const int row0  = mt << 4;
    const int mrow  = lane & 15;
    const int khalf = (lane >> 4) << 1;          // 0 or 2
    const int col   = (nt << 4) + mrow;          // may be >= N (reads zeros)

    const float*  Arow  = A + (size_t)(row0 + mrow) * K;
    const float2* BTrow = (const float2*)(BT + col * KS);

    v8f acc = {};
    constexpr int KFULL = K & ~3;
#pragma unroll
    for (int k0 = 0; k0 < KFULL; k0 += 4) {
        const int ka = k0 + khalf;
        const float2 av = *(const float2*)(Arow + ka);   // global_load_b64
        const float2 bv = BTrow[ka >> 1];                // ds_load_b64
        v2f a; a.x = av.x; a.y = av.y;
        v2f b; b.x = bv.x; b.y = bv.y;
        acc = __builtin_amdgcn_wmma_f32_16x16x4_f32(
            false, a, false, b, (short)0, acc, false, false);
    }
    if constexpr (KFULL < K) {
        // Single ragged-K tail step (only lanes whose ka >= K load zeros).
        const int ka = KFULL + khalf;
        v2f a;
        a.x = (ka     < K) ? Arow[ka]     : 0.0f;
        a.y = (ka + 1 < K) ? Arow[ka + 1] : 0.0f;
        const float2 bv = BTrow[ka >> 1];                // zero-padded in LDS
        v2f b; b.x = bv.x; b.y = bv.y;
        acc = __builtin_amdgcn_wmma_f32_16x16x4_f32(
            false, a, false, b, (short)0, acc, false, false);
    }

    if (col < N) {
        const int rbase = row0 + ((lane >> 4) << 3);
#pragma unroll
        for (int r = 0; r < 8; ++r)
            C[(size_t)(rbase + r) * N + col] = acc[r];
    }
}

// ---------------------------------------------------------------------------
// Edge scatter: acc[dst[e], :] += dinv[src]*dinv[dst] * xw[src[e], :]
// One wave32 per edge, lanes stride features; atomics resolve in L2.
// ---------------------------------------------------------------------------
__global__ void gcn_scatter_kernel(const float* __restrict__ xw,
                                   const int* __restrict__ src,
                                   const int* __restrict__ dst,
                                   const float* __restrict__ dinv,
                                   float* __restrict__ acc,
                                   int nE, int F) {
    const int wavesPerBlock = blockDim.x >> 5;
    const int e = blockIdx.x * wavesPerBlock + (threadIdx.x >> 5);
    if (e >= nE) return;
    const int lane = threadIdx.x & 31;
    const int s = src[e];
    const int d = dst[e];
    const float norm = dinv[s] * dinv[d];
    const float* xs = xw + (size_t)s * F;
    float* ad = acc + (size_t)d * F;
    for (int f = lane; f < F; f += 32)
        atomicAdd(&ad[f], norm * xs[f]);
}

// ---------------------------------------------------------------------------
// Self-loop message + optional tanh:
//   out[i,f] = act( acc[i,f] + dinv[i]^2 * xw[i,f] )   (out may alias acc)
// ---------------------------------------------------------------------------
__global__ void gcn_selfloop_kernel(const float* __restrict__ acc,
                                    const float* __restrict__ xw,
                                    const float* __restrict__ dinv,
                                    float* __restrict__ out,
                                    int nNodes, int F, int doTanh) {
    int idx = blockIdx.x * blockDim.x + threadIdx.x;
    int total = nNodes * F;
    int stride = gridDim.x * blockDim.x;
    for (; idx < total; idx += stride) {
        const int node = idx / F;
        const float di = dinv[node];
        float v = acc[idx] + di * di * xw[idx];
        out[idx] = doTanh ? tanhf(v) : v;
    }
}

// ---------------------------------------------------------------------------
// Host-side launch plumbing
// ---------------------------------------------------------------------------
extern "C" void kernel_launch(void* const* d_in, const int* in_sizes, int n_in,
                              void* d_out, int out_size, void* d_ws, size_t ws_size,
                              hipStream_t stream) {
    const float* x  = (const float*)d_in[0];   // [N, 50]
    const float* W1 = (const float*)d_in[1];   // [50, 64]
    const float* W2 = (const float*)d_in[2];   // [64, 121]
    const int* ei   = (const int*)d_in[3];     // [2, E]

    const int FIN = 50, F1 = 64, F2 = 121;
    const int nNodes = in_sizes[0] / FIN;      // 50000
    const int nEdges = in_sizes[3] / 2;        // 800000
    const int* src = ei;
    const int* dst = ei + nEdges;

    // Workspace layout (floats)
    float* wsF = (float*)d_ws;
    float* dinv = wsF;                                   // nNodes
    float* xw1  = dinv + ((nNodes + 63) & ~63);          // nNodes * F1
    float* hbuf = xw1 + (size_t)nNodes * F1;             // nNodes * F1 (acc -> h)
    float* xw2  = hbuf + (size_t)nNodes * F1;            // nNodes * F2
    float* outf = (float*)d_out;                         // nNodes * F2

    const int T = 256;
    const int wavesPerBlock = T / 32;

    // --- normalization: deg = 1 + in-degree ; dinv = rsqrt(deg) ---
    gcn_fill_kernel<<<(nNodes + T - 1) / T, T, 0, stream>>>(dinv, 1.0f, nNodes);
    gcn_degree_kernel<<<(nEdges + T - 1) / T, T, 0, stream>>>(dst, dinv, nEdges);
    gcn_rsqrt_kernel<<<(nNodes + T - 1) / T, T, 0, stream>>>(dinv, nNodes);

    // --- layer 1: xw1 = x @ W1 (WMMA f32, K=50, N=64) ---
    {
        const int tiles = (nNodes / 16) * 4;             // 12500
        const int blocks = (tiles + wavesPerBlock - 1) / wavesPerBlock;
        gcn_gemm_wmma_kernel<50, 64, 4><<<blocks, T, 0, stream>>>(x, W1, xw1, nNodes);
    }
    // scatter + self loop + tanh -> h (in hbuf)
    gcn_fill_kernel<<<4096, T, 0, stream>>>(hbuf, 0.0f, nNodes * F1);
    gcn_scatter_kernel<<<(nEdges + wavesPerBlock - 1) / wavesPerBlock, T, 0, stream>>>(
        xw1, src, dst, dinv, hbuf, nEdges, F1);
    gcn_selfloop_kernel<<<4096, T, 0, stream>>>(hbuf, xw1, dinv, hbuf, nNodes, F1, 1);

    // --- layer 2: xw2 = h @ W2 (WMMA f32, K=64, N=121) ---
    {
        const int tiles = (nNodes / 16) * 8;             // 25000
        const int blocks = (tiles + wavesPerBlock - 1) / wavesPerBlock;
        gcn_gemm_wmma_kernel<64, 121, 8><<<blocks, T, 0, stream>>>(hbuf, W2, xw2, nNodes);
    }
    // scatter into d_out + self loop (no tanh)
    gcn_fill_kernel<<<4096, T, 0, stream>>>(outf, 0.0f, nNodes * F2);
    gcn_scatter_kernel<<<(nEdges + wavesPerBlock - 1) / wavesPerBlock, T, 0, stream>>>(
        xw2, src, dst, dinv, outf, nEdges, F2);
    gcn_selfloop_kernel<<<4096, T, 0, stream>>>(outf, xw2, dinv, outf, nNodes, F2, 0);

    (void)n_in; (void)out_size; (void)ws_size;
}